// DepthLoss_16810501997336
// MI455X (gfx1250) — compile-verified
//
#include <hip/hip_runtime.h>

// ---------------------------------------------------------------------------
// DepthLoss for MI455X / gfx1250 (wave32).
// B=32, N=16384, H=768, W=1024 (fixed by the reference).
// Pipeline: async-copy rdepth tiles to LDS (ASYNCcnt), random-gather from the
// image (L2-resident: 100MB < 192MB L2), per-thread f32 partials, WMMA-based
// block reduction, two global f32 atomics per block, final scalar divide.
// ---------------------------------------------------------------------------

#define TPB   256
#define TILE  256                  // items per tile (one item = 3 f32 = 12B)
#define TILEF (TILE * 3)           // 768 floats = 3072 bytes per tile

typedef float v2f __attribute__((ext_vector_type(2)));
typedef float v8f __attribute__((ext_vector_type(8)));
typedef int   v4i __attribute__((vector_size(4 * sizeof(int))));

typedef __attribute__((address_space(1))) v4i gv4i;   // global
typedef __attribute__((address_space(3))) v4i lv4i;   // LDS

#if __has_builtin(__builtin_amdgcn_global_load_async_to_lds_b128)
#define HAVE_ASYNC_LDS 1
#else
#define HAVE_ASYNC_LDS 0
#endif

#if __has_builtin(__builtin_amdgcn_wmma_f32_16x16x4_f32)
#define HAVE_WMMA_F32X4 1
#else
#define HAVE_WMMA_F32X4 0
#endif

__device__ __forceinline__ void dl_wait_async(int n) {
#if __has_builtin(__builtin_amdgcn_s_wait_asynccnt)
  if (n == 0) __builtin_amdgcn_s_wait_asynccnt(0);
  else        __builtin_amdgcn_s_wait_asynccnt(1);
#else
  if (n == 0) asm volatile("s_wait_asynccnt 0x0" ::: "memory");
  else        asm volatile("s_wait_asynccnt 0x1" ::: "memory");
#endif
}

// Issue the async copy of tile t into tile_buf. 192 lanes x 16B = 3072B.
// Lanes 192..255 (waves 6,7) issue nothing; a later barrier covers them.
__device__ __forceinline__ void dl_issue_tile(const float* __restrict__ rdepth,
                                              int t, float* __restrict__ dstTile,
                                              int tid) {
  if (tid < 192) {
    const float* gsrc = rdepth + (size_t)t * TILEF + (size_t)tid * 4;
    float*       ldst = dstTile + tid * 4;
#if HAVE_ASYNC_LDS
    __builtin_amdgcn_global_load_async_to_lds_b128(
        (gv4i*)gsrc, (lv4i*)ldst, 0 /*imm offset*/, 0 /*cpol*/);
#else
    *(float4*)ldst = *(const float4*)gsrc;
#endif
  }
}

#if HAVE_WMMA_F32X4
// Sum 256 f32 values in LDS using V_WMMA_F32_16X16X4_F32 with B = ones:
// D[m][n] = C[m][n] + sum_k A[m][k].  Accumulate 4 chunks of 64 values; the
// total is sum_m rowsum(m) = (lane0 + lane16) of the per-lane 8-VGPR sums.
// Must be executed by a full wave (EXEC all ones).
__device__ __forceinline__ float dl_wmma_sum256(const float* s, int lane) {
  v8f acc = {0.f, 0.f, 0.f, 0.f, 0.f, 0.f, 0.f, 0.f};
  v2f ones;
  ones[0] = 1.0f;
  ones[1] = 1.0f;
#pragma unroll
  for (int ch = 0; ch < 4; ++ch) {
    v2f a;
    a[0] = s[ch * 64 + lane];
    a[1] = s[ch * 64 + 32 + lane];
    acc = __builtin_amdgcn_wmma_f32_16x16x4_f32(
        /*neg_a=*/false, a, /*neg_b=*/false, ones,
        /*c_mod=*/(short)0, acc, /*reuse_a=*/false, /*reuse_b=*/false);
  }
  float per = 0.f;
#pragma unroll
  for (int i = 0; i < 8; ++i) per += acc[i];
  return __shfl(per, 0, 32) + __shfl(per, 16, 32);
}
#else
// Fallback: strided gather + shuffle tree (wave32).
__device__ __forceinline__ float dl_wmma_sum256(const float* s, int lane) {
  float v = 0.f;
#pragma unroll
  for (int k = 0; k < 8; ++k) v += s[lane + 32 * k];
#pragma unroll
  for (int off = 16; off > 0; off >>= 1) v += __shfl_xor(v, off, 32);
  return v;
}
#endif

__global__ void __launch_bounds__(TPB)
depth_loss_main(const float* __restrict__ out2d,   // B*H*W
                const float* __restrict__ rdepth,  // B*N*3
                float* __restrict__ acc,           // [0]=loss, [1]=count
                int numTiles) {
  constexpr int W = 1024;
  constexpr int HW = 768 * 1024;

  __shared__ __align__(16) float tile[2][TILEF];
  __shared__ float sred[2 * TPB];

  const int tid = threadIdx.x;
  const int stride = gridDim.x;

  float loss = 0.f;
  float cnt = 0.f;

  int t = blockIdx.x;
  int buf = 0;
  if (t < numTiles) dl_issue_tile(rdepth, t, &tile[0][0], tid);

  for (; t < numTiles; t += stride, buf ^= 1) {
    const int tn = t + stride;
    // Prefetch next tile into the other buffer (free: end-of-iteration
    // barrier of the previous loop pass guarantees nobody is reading it).
    if (tn < numTiles) dl_issue_tile(rdepth, tn, &tile[buf ^ 1][0], tid);
    dl_wait_async(tn < numTiles ? 1 : 0);  // oldest (current tile) complete
    __syncthreads();                       // visible to all 8 waves

    // Process one item per thread. Stride-3 LDS reads: gcd(3,64)=1 -> no
    // bank conflicts.
    const float rr = tile[buf][tid * 3 + 0];
    const float cc = tile[buf][tid * 3 + 1];
    const float dd = tile[buf][tid * 3 + 2];
    const int idx = t * TILE + tid;
    const int b = idx >> 14;               // idx / N, N = 16384
    if (dd > 0.f) {
      const int r = (int)rr;
      const int c = (int)cc;
      const float v = out2d[(size_t)b * HW + (size_t)r * W + c];
      loss += fabsf(v - dd);
      cnt += 1.f;
    }
    __syncthreads();                       // done reading tile[buf]
  }

  // Block reduction of 256 partials via WMMA (wave 0 only, EXEC all-ones).
  sred[tid] = loss;
  sred[TPB + tid] = cnt;
  __syncthreads();
  if (tid < 32) {
    const float L = dl_wmma_sum256(&sred[0], tid);
    const float C = dl_wmma_sum256(&sred[TPB], tid);
    if (tid == 0) {
      atomicAdd(&acc[0], L);
      atomicAdd(&acc[1], C);
    }
  }
}

__global__ void depth_loss_init(float* __restrict__ acc) {
  if (threadIdx.x == 0) {
    acc[0] = 0.f;
    acc[1] = 0.f;
  }
}

__global__ void depth_loss_final(const float* __restrict__ acc,
                                 float* __restrict__ out) {
  if (threadIdx.x == 0) {
    const float L = acc[0];
    const float C = acc[1];
    out[0] = (C > 0.f) ? (L / C) : 0.f;
  }
}

extern "C" void kernel_launch(void* const* d_in, const int* in_sizes, int n_in,
                              void* d_out, int out_size, void* d_ws,
                              size_t ws_size, hipStream_t stream) {
  (void)in_sizes; (void)n_in; (void)out_size; (void)ws_size;
  const float* out2d  = (const float*)d_in[0];  // (B,1,H,W) f32
  const float* rdepth = (const float*)d_in[1];  // (B,N,3)   f32
  float* acc = (float*)d_ws;                    // 2 floats of scratch
  float* out = (float*)d_out;                   // 1 float

  constexpr int totalItems = 32 * 16384;        // B*N
  constexpr int numTiles = totalItems / TILE;   // 2048
  const int blocks = (numTiles < 1024) ? numTiles : 1024;

  depth_loss_init<<<1, 64, 0, stream>>>(acc);
  depth_loss_main<<<blocks, TPB, 0, stream>>>(out2d, rdepth, acc, numTiles);
  depth_loss_final<<<1, 64, 0, stream>>>(acc, out);
}